// RoadBlockConsistencyLoss_4844723110183
// MI455X (gfx1250) — compile-verified
//
#include <hip/hip_runtime.h>
#include <math.h>

#define NBLK      100
#define NBLK_PAD  112      // padded so WMMA tiles (7 x 16) read zeroed rows past 100
#define DIMS      128
#define EPSV      1e-8f

typedef float v2f __attribute__((ext_vector_type(2)));
typedef float v8f __attribute__((ext_vector_type(8)));

// Workspace layout (floats): cnt[112] | sums[112*128] | S[112*128]
#define WS_CNT 0
#define WS_SUM (NBLK_PAD)
#define WS_S   (NBLK_PAD + NBLK_PAD * DIMS)
#define WS_TOT (NBLK_PAD + 2 * NBLK_PAD * DIMS)

__global__ void rbc_init_kernel(float* __restrict__ acc) {
    int i = blockIdx.x * blockDim.x + threadIdx.x;
    if (i < WS_TOT) acc[i] = 0.0f;
}

// ---------------------------------------------------------------------------
// Pass 1: single streaming pass over z (512 MB).  Wave-per-point: lane L loads
// float4 = dims [4L..4L+3] (one coalesced b128 per lane).  Per point we
// accumulate z and z/max(|z|,eps) into LDS-private per-block arrays using a
// bank-conflict-free permuted layout (dim 4L+j stored at slot L+32j: banks
// L+32j mod 64 are distinct across the 32 lanes for each ds_add_f32).
// PTS=8 points in flight per wave for memory-level parallelism
// (~128 WGs * 32 waves * 8 * 512B ~= 17 MB outstanding -> saturates 23.3 TB/s).
// ---------------------------------------------------------------------------
__global__ __launch_bounds__(1024) void rbc_accum_kernel(
    const float* __restrict__ z, const int* __restrict__ ids,
    float* __restrict__ acc, int N) {
    __shared__ float lsum[NBLK * DIMS];   //  51.2 KB
    __shared__ float lS[NBLK * DIMS];     //  51.2 KB
    __shared__ float lcnt[NBLK_PAD];      // 103 KB total (<320 KB/WGP on CDNA5)

    const int tid = threadIdx.x;
    for (int i = tid; i < NBLK * DIMS; i += blockDim.x) { lsum[i] = 0.f; lS[i] = 0.f; }
    if (tid < NBLK_PAD) lcnt[tid] = 0.f;
    __syncthreads();

    const int lane = tid & 31;
    const int wave = tid >> 5;
    const int wavesPerWG = blockDim.x >> 5;
    const long long wid = (long long)blockIdx.x * wavesPerWG + wave;
    const long long W   = (long long)gridDim.x * wavesPerWG;
    const int PTS = 8;

    for (long long base = wid * PTS; base < N; base += W * PTS) {
        float4 v[PTS];
        int    b[PTS];
#pragma unroll
        for (int p = 0; p < PTS; ++p) {
            long long i = base + p;
            if (i < N) {
                const float4* row = (const float4*)(z + (size_t)i * DIMS);
                v[p] = row[lane];          // global_load_b128, coalesced 512B/wave
                b[p] = ids[i];             // wave-uniform
            } else {
                v[p] = make_float4(0.f, 0.f, 0.f, 0.f);
                b[p] = -1;
            }
        }
#pragma unroll
        for (int p = 0; p < PTS; ++p) {
            if (b[p] < 0) continue;        // wave-uniform branch
            float4 x = v[p];
            float s = x.x * x.x + x.y * x.y + x.z * x.z + x.w * x.w;
            for (int m = 16; m >= 1; m >>= 1) s += __shfl_xor(s, m, 32);
            float inv = 1.0f / fmaxf(sqrtf(s), EPSV);   // 1/max(|z|,eps)
            int off = b[p] * DIMS + lane;  // permuted slot: dim 4L+j -> L+32j
            atomicAdd(&lsum[off     ], x.x);
            atomicAdd(&lsum[off + 32], x.y);
            atomicAdd(&lsum[off + 64], x.z);
            atomicAdd(&lsum[off + 96], x.w);
            atomicAdd(&lS[off     ], x.x * inv);
            atomicAdd(&lS[off + 32], x.y * inv);
            atomicAdd(&lS[off + 64], x.z * inv);
            atomicAdd(&lS[off + 96], x.w * inv);
            if (lane == 0) atomicAdd(&lcnt[b[p]], 1.0f);
        }
    }
    __syncthreads();

    // Flush LDS partials to global accumulators (same permuted layout).
    float* gsum = acc + WS_SUM;
    float* gS   = acc + WS_S;
    for (int i = tid; i < NBLK * DIMS; i += blockDim.x) {
        atomicAdd(&gsum[i], lsum[i]);
        atomicAdd(&gS[i],   lS[i]);
    }
    if (tid < NBLK) atomicAdd(&acc[WS_CNT + tid], lcnt[tid]);
}

// ---------------------------------------------------------------------------
// Pass 2: per-block  d1 = S_b . sums_b  and  d2 = sums_b . sums_b  via
// V_WMMA_F32_16X16X4_F32 (f32, exact).  One wave per 16-block tile (7 tiles,
// padded rows 100..111 are zero).  A layout (ISA 7.12.2, 32-bit A 16x4):
// row M = lane&15 for both lane halves, K = 2*(lane>>4) + vgpr.  B symmetric
// with column N = lane&15.  Diagonal of D (32-bit C/D layout): lanes 0..7 hold
// rows 0..7 in accumulator register lane; lanes 24..31 hold rows 8..15 in register lane-24.
// Dot products are invariant under the (shared) dim permutation from pass 1.
// ---------------------------------------------------------------------------
__device__ __forceinline__ float rbc_pick8(v8f c, int r) {
    switch (r & 7) {
        case 0: return c[0]; case 1: return c[1]; case 2: return c[2]; case 3: return c[3];
        case 4: return c[4]; case 5: return c[5]; case 6: return c[6]; default: return c[7];
    }
}

__global__ __launch_bounds__(256) void rbc_finalize_kernel(
    const float* __restrict__ acc, float* __restrict__ out) {
    __shared__ float d1s[NBLK_PAD];
    __shared__ float d2s[NBLK_PAD];
    __shared__ float terms[NBLK];
    __shared__ float vlds[NBLK];

    const int tid  = threadIdx.x;
    const int wave = tid >> 5;
    const int lane = tid & 31;
    const float* gcnt = acc + WS_CNT;
    const float* gsum = acc + WS_SUM;
    const float* gS   = acc + WS_S;

    if (wave < 7) {                      // 7 tiles of 16 blocks cover 112 rows
        const int t = wave;
        const int g = lane >> 4;         // lane half selects K pair
        const int l = lane & 15;         // A row / B column within tile
        const float* Srow = gS   + (size_t)(16 * t + l) * DIMS;
        const float* Mrow = gsum + (size_t)(16 * t + l) * DIMS;
        v8f c1 = {};                     // accumulates S x sums^T
        v8f c2 = {};                     // accumulates sums x sums^T (Gram)
#pragma unroll 4
        for (int k = 0; k < DIMS; k += 4) {
            const int k0 = k + 2 * g;
            v2f a1; a1[0] = Srow[k0]; a1[1] = Srow[k0 + 1];
            v2f bm; bm[0] = Mrow[k0]; bm[1] = Mrow[k0 + 1];
            c1 = __builtin_amdgcn_wmma_f32_16x16x4_f32(
                     false, a1, false, bm, (short)0, c1, false, false);
            c2 = __builtin_amdgcn_wmma_f32_16x16x4_f32(
                     false, bm, false, bm, (short)0, c2, false, false);
        }
        int r = -1, blk = -1;
        if (lane < 8)        { r = lane;      blk = 16 * t + lane;        }
        else if (lane >= 24) { r = lane - 24; blk = 16 * t + (lane - 16); }
        if (r >= 0) {
            d1s[blk] = rbc_pick8(c1, r);
            d2s[blk] = rbc_pick8(c2, r);
        }
    }
    __syncthreads();

    if (tid < NBLK) {
        float cnt = gcnt[tid];
        float M   = fmaxf(cnt, 1.0f);
        float cn  = fmaxf(sqrtf(fmaxf(d2s[tid], 0.0f)) / M, EPSV); // c_norm
        float mean_cos = (d1s[tid] / M) / (cn * M);
        float valid = (cnt > 1.0f) ? 1.0f : 0.0f;
        terms[tid] = valid * (1.0f - mean_cos);
        vlds[tid]  = valid;
    }
    __syncthreads();
    if (tid == 0) {
        float tot = 0.f, c = 0.f;
        for (int i = 0; i < NBLK; ++i) { tot += terms[i]; c += vlds[i]; }
        out[0] = tot / fmaxf(c, 1.0f);
    }
}

extern "C" void kernel_launch(void* const* d_in, const int* in_sizes, int n_in,
                              void* d_out, int out_size, void* d_ws, size_t ws_size,
                              hipStream_t stream) {
    const float* z   = (const float*)d_in[0];
    const int*   ids = (const int*)d_in[1];
    const int N = in_sizes[1];            // z is [N,128], ids is [N]
    float* acc = (float*)d_ws;            // needs WS_TOT*4 = ~115 KB
    float* out = (float*)d_out;

    rbc_init_kernel<<<(WS_TOT + 255) / 256, 256, 0, stream>>>(acc);
    rbc_accum_kernel<<<128, 1024, 0, stream>>>(z, ids, acc, N);
    rbc_finalize_kernel<<<1, 256, 0, stream>>>(acc, out);
    (void)n_in; (void)out_size; (void)ws_size;
}